// CTCLayer_64381559767091
// MI455X (gfx1250) — compile-verified
//
#include <hip/hip_runtime.h>
#include <hip/hip_bf16.h>
#include <stdint.h>

// Problem constants (match the reference)
#define B_   128
#define T_   1024
#define C_   256
#define L_   128
#define S_   257          // 2L+1 extended label length
#define BLK  288          // 9 wave32s; threads [0,257) own one s-position
#define D_   4            // async pipeline depth (power of two)
#define NEGV (-1e30f)
#define EPSV (1e-7f)

// ---- CDNA5 async global->LDS path (guarded; falls back to sync loads) ----
#if defined(__has_builtin)
#  if __has_builtin(__builtin_amdgcn_global_load_async_to_lds_b128) && \
      __has_builtin(__builtin_amdgcn_s_wait_asynccnt)
#    define USE_ASYNC 1
#  endif
#endif
#ifndef USE_ASYNC
#  define USE_ASYNC 0
#endif

typedef __attribute__((__vector_size__(16))) int v4i;
typedef __attribute__((address_space(1))) v4i gv4i;   // global (prints as __device__)
typedef __attribute__((address_space(3))) v4i lv4i;   // LDS    (prints as __shared__)

#if USE_ASYNC
__device__ __forceinline__ void async_copy16(const float* g, float* l) {
    // GLOBAL_LOAD_ASYNC_TO_LDS_B128: 16 bytes per active lane, tracked by ASYNCcnt
    __builtin_amdgcn_global_load_async_to_lds_b128((gv4i*)g, (lv4i*)l, 0, 0);
}
#endif

__device__ __forceinline__ float lae(float a, float b) {
    // logaddexp; NEGV acts as log-space zero (exp underflows to 0)
    float m = fmaxf(a, b);
    float d = fabsf(a - b);
    return m + log1pf(__expf(-d));
}

__global__ __launch_bounds__(BLK) void ctc_fwd_kernel(
        const int* __restrict__ y_true,     // [B, L]
        const float* __restrict__ y_pred,   // [B, T, C] softmax probs
        float* __restrict__ out)            // [B, 1]
{
    __shared__ float probBuf[D_ * C_];      // 4 KB ring of prob rows
    __shared__ float alphaBuf[2][S_ + 2];   // ping-pong, 2-elem NEG front pad

    const int b   = blockIdx.x;
    const int tid = threadIdx.x;
    const float* rowBase = y_pred + (size_t)b * T_ * C_;

    // Per-thread extended-label state (registers)
    int  ext_s = C_ - 1;                    // blank
    bool allow = false;
    if (tid < S_) {
        if (tid & 1) {
            const int j = tid >> 1;
            ext_s = y_true[b * L_ + j];
            allow = (tid >= 3) && (ext_s != y_true[b * L_ + j - 1]);
        }
    }

    // ---- Prefetch rows 0..D-1 into the LDS ring ----
#if USE_ASYNC
    if (tid < 64) {
        #pragma unroll
        for (int k = 0; k < D_; ++k) {
            async_copy16(rowBase + k * C_ + tid * 4,
                         &probBuf[k * C_ + tid * 4]);
        }
        __builtin_amdgcn_s_wait_asynccnt(D_ - 1);            // row 0 landed
    }
#else
    if (tid < C_) probBuf[tid] = rowBase[tid];
#endif
    __syncthreads();

    // ---- alpha(t=0): only s=0 (blank) and s=1 (first label) reachable ----
    if (tid < 2) { alphaBuf[0][tid] = NEGV; alphaBuf[1][tid] = NEGV; }
    if (tid < S_) {
        float a0 = NEGV;
        if (tid < 2) a0 = __logf(probBuf[ext_s] + EPSV);
        alphaBuf[0][tid + 2] = a0;
    }
    __syncthreads();                         // row-0 reads + alpha0 writes done
#if USE_ASYNC
    if (tid < 64) {                          // issue row D into freed slot 0
        async_copy16(rowBase + D_ * C_ + tid * 4, &probBuf[tid * 4]);
    }
#endif

    int cur = 0;
    for (int t = 1; t < T_; ++t) {
#if USE_ASYNC
        const int bi = (t & (D_ - 1)) * C_;
        if (tid < 64) {
            if (t <= T_ - D_) __builtin_amdgcn_s_wait_asynccnt(D_ - 1); // row t done
            else              __builtin_amdgcn_s_wait_asynccnt(0);     // drain tail
        }
        __syncthreads();                     // A: row t visible; alpha(t-1) visible
#else
        const int bi = 0;
        __syncthreads();                     // A: previous reads done
        if (tid < C_) probBuf[tid] = rowBase[(size_t)t * C_ + tid];
        __syncthreads();                     // row t visible
#endif
        float anew = 0.0f;
        if (tid < S_) {
            const float p  = probBuf[bi + ext_s];
            const float lp = __logf(p + EPSV);
            const float a  = alphaBuf[cur][tid + 2];
            const float a1 = alphaBuf[cur][tid + 1];
            const float a2 = allow ? alphaBuf[cur][tid] : NEGV;
            anew = lae(lae(a, a1), a2) + lp;
        }
        __syncthreads();                     // B: ring slot + alpha(cur) fully read
        if (tid < S_) alphaBuf[cur ^ 1][tid + 2] = anew;
#if USE_ASYNC
        if (tid < 64 && (t + D_) < T_) {     // refill the slot just consumed
            async_copy16(rowBase + (size_t)(t + D_) * C_ + tid * 4,
                         &probBuf[bi + tid * 4]);
        }
#endif
        cur ^= 1;
    }

    __syncthreads();
    if (tid == 0) {
        // end in last blank (s=S-1 -> idx S_+1) or last label (s=S-2 -> idx S_)
        const float ll = lae(alphaBuf[cur][S_ + 1], alphaBuf[cur][S_]);
        out[b] = -ll;
    }
}

extern "C" void kernel_launch(void* const* d_in, const int* in_sizes, int n_in,
                              void* d_out, int out_size, void* d_ws, size_t ws_size,
                              hipStream_t stream) {
    (void)in_sizes; (void)n_in; (void)out_size; (void)d_ws; (void)ws_size;
    const int*   y_true = (const int*)d_in[0];
    const float* y_pred = (const float*)d_in[1];
    float*       out    = (float*)d_out;
    ctc_fwd_kernel<<<dim3(B_), dim3(BLK), 0, stream>>>(y_true, y_pred, out);
}